// BimanualSpatialAttention_17308718203176
// MI455X (gfx1250) — compile-verified
//
#include <hip/hip_runtime.h>

typedef float v2f __attribute__((ext_vector_type(2)));
typedef float v8f __attribute__((ext_vector_type(8)));

#define T_LEN   4096
#define CHUNK   256
#define HALO    2
#define XROW    (CHUNK + HALO)   // 258 token rows staged in LDS
#define XSTR    20               // floats per token row (16 + pad), 80B => 16B aligned rows
#define HCSTR   34               // H_cat scratch stride (conflict-free 32-lane stores)
#define HDSTR   20               // hidden scratch stride (16B-aligned b128 reads)
#define NWAVE   8

__global__ __launch_bounds__(256) void
bimanual_fused(const float* __restrict__ x,
               const float* __restrict__ conv_w,   // (16,8,3)
               const float* __restrict__ conv_b,   // (16,)
               const float* __restrict__ w1,       // (32,16)
               const float* __restrict__ b1,       // (16,)
               const float* __restrict__ w2,       // (16,2)
               const float* __restrict__ b2,       // (2,)
               const float* __restrict__ ln_g,     // (16,)
               const float* __restrict__ ln_b,     // (16,)
               float* __restrict__ out_x,
               float* __restrict__ out_a)
{
  __shared__ __align__(16) float lds_x[XROW * XSTR];          // staged x chunk (+halo)
  __shared__ __align__(16) float lds_hc[NWAVE * 16 * HCSTR];  // per-wave H_cat transpose scratch
  __shared__ __align__(16) float lds_hd[NWAVE * 32 * HDSTR];  // per-wave hidden transpose scratch
  __shared__ __align__(16) float lds_par[64];                 // w2diff[16], ln_g[16], ln_b[16], b2diff

  const int tid  = threadIdx.x;
  const int lane = tid & 31;
  const int wave = tid >> 5;
  const int row   = blockIdx.x >> 4;        // batch index (16 chunks per row of T=4096)
  const int t0    = (blockIdx.x & 15) * CHUNK;
  const long long x_base = (long long)row * T_LEN * 16;

  // ---- stage x chunk (with zero-filled causal halo) into LDS, coalesced
  for (int i = tid; i < XROW * 16; i += 256) {
    int tt = i >> 4;              // 0..257  -> global token t0 + tt - 2
    int cc = i & 15;
    int tg = t0 + tt - HALO;
    float v = 0.0f;
    if (tg >= 0) v = x[x_base + (long long)tg * 16 + cc];
    lds_x[tt * XSTR + cc] = v;
  }
  if (tid < 16) {
    lds_par[tid]      = w2[tid * 2 + 0] - w2[tid * 2 + 1];
    lds_par[16 + tid] = ln_g[tid];
    lds_par[32 + tid] = ln_b[tid];
  }
  if (tid == 0) lds_par[48] = b2[0] - b2[1];

  // ---- constant B-layout operands in registers (per-lane gathers, L2-resident)
  const int nlo = lane & 15;
  const int hi2 = (lane >> 4) << 1;    // 0 for lanes 0-15, 2 for lanes 16-31

  // conv as (24 x 16) matmul: K = tap*8 + ch ; B[K][n] = conv_w[n, ch, tap]
  v2f Bc[6];
#pragma unroll
  for (int j = 0; j < 6; ++j) {
    int K0 = 4 * j + hi2;
    int K1 = K0 + 1;
    Bc[j].x = conv_w[nlo * 24 + (K0 & 7) * 3 + (K0 >> 3)];
    Bc[j].y = conv_w[nlo * 24 + (K1 & 7) * 3 + (K1 >> 3)];
  }
  // w1 (32 x 16): B[K][n] = w1[K*16 + n]
  v2f Bw1[8];
#pragma unroll
  for (int j = 0; j < 8; ++j) {
    int K0 = 4 * j + hi2;
    Bw1[j].x = w1[K0 * 16 + nlo];
    Bw1[j].y = w1[(K0 + 1) * 16 + nlo];
  }
  const float vcb = conv_b[nlo];
  const float vb1 = b1[nlo];

  __syncthreads();

  float* my_hc = &lds_hc[wave * 16 * HCSTR];
  float* my_hd = &lds_hd[wave * 32 * HDSTR];
  const int wbase = wave * 32;             // this wave's 32 tokens within the chunk
  const int m     = lane & 15;             // M index (token within 16-token tile)
  const int rhalf = (lane >> 4) << 3;      // C/D-layout row offset: 0 or 8

#pragma unroll
  for (int sub = 0; sub < 2; ++sub) {
    const int base = wbase + sub * 16;

    // ---- matmul1: H_L / H_R tiles (16 tok x 16 ch), K = 24 each, f32 WMMA
    v8f cL = {}; v8f cR = {};
#pragma unroll
    for (int j = 0; j < 6; ++j) {
      int K0 = 4 * j + hi2;
      int tap = K0 >> 3, ch = K0 & 7;
      const float* pa = &lds_x[(base + m + tap) * XSTR + ch];
      v2f a; a.x = pa[0]; a.y = pa[1];
      cL = __builtin_amdgcn_wmma_f32_16x16x4_f32(false, a, false, Bc[j],
                                                 (short)0, cL, false, false);
    }
#pragma unroll
    for (int j = 0; j < 6; ++j) {
      int K0 = 4 * j + hi2;
      int tap = K0 >> 3, ch = (K0 & 7) + 8;
      const float* pa = &lds_x[(base + m + tap) * XSTR + ch];
      v2f a; a.x = pa[0]; a.y = pa[1];
      cR = __builtin_amdgcn_wmma_f32_16x16x4_f32(false, a, false, Bc[j],
                                                 (short)0, cR, false, false);
    }

    // bias + relu ; transpose H_cat tile through per-wave LDS scratch
#pragma unroll
    for (int g = 0; g < 8; ++g) {
      float hl = fmaxf(cL[g] + vcb, 0.0f);
      float hr = fmaxf(cR[g] + vcb, 0.0f);
      my_hc[(rhalf + g) * HCSTR + nlo]      = hl;   // channels 0..15  (L)
      my_hc[(rhalf + g) * HCSTR + 16 + nlo] = hr;   // channels 16..31 (R)
    }

    // ---- matmul2: hidden = relu(H_cat @ w1 + b1), K = 32
    v8f c2 = {};
#pragma unroll
    for (int j = 0; j < 8; ++j) {
      int K0 = 4 * j + hi2;
      const float* pa = &my_hc[m * HCSTR + K0];
      v2f a; a.x = pa[0]; a.y = pa[1];
      c2 = __builtin_amdgcn_wmma_f32_16x16x4_f32(false, a, false, Bw1[j],
                                                 (short)0, c2, false, false);
    }
#pragma unroll
    for (int g = 0; g < 8; ++g) {
      float hv = fmaxf(c2[g] + vb1, 0.0f);
      my_hd[(sub * 16 + rhalf + g) * HDSTR + nlo] = hv;
    }
  }

  // ---- epilogue: one token per lane (32 tokens per wave)
  {
    const int  tt = lane;
    const long long tok = (long long)row * T_LEN + t0 + wbase + tt;

    float hid[16];
#pragma unroll
    for (int q = 0; q < 4; ++q) {
      float4 v = *(const float4*)&my_hd[tt * HDSTR + q * 4];
      hid[q*4+0] = v.x; hid[q*4+1] = v.y; hid[q*4+2] = v.z; hid[q*4+3] = v.w;
    }
    float d = lds_par[48];
#pragma unroll
    for (int c = 0; c < 16; ++c) d = fmaf(hid[c], lds_par[c], d);
    float a0 = 1.0f / (1.0f + expf(-d));   // softmax([l0,l1])[0] == sigmoid(l0-l1), exact
    float a1 = 1.0f / (1.0f + expf(d));

    float xv[16];
#pragma unroll
    for (int q = 0; q < 4; ++q) {
      float4 v = *(const float4*)&lds_x[(wbase + tt + HALO) * XSTR + q * 4];
      xv[q*4+0] = v.x; xv[q*4+1] = v.y; xv[q*4+2] = v.z; xv[q*4+3] = v.w;
    }
    float y[16]; float mu = 0.0f;
#pragma unroll
    for (int c = 0; c < 16; ++c) {
      float s = (c < 8) ? a0 : a1;
      y[c] = xv[c] * (1.0f + s);           // x + alpha*x
      mu += y[c];
    }
    mu *= (1.0f / 16.0f);
    float var = 0.0f;
#pragma unroll
    for (int c = 0; c < 16; ++c) { float t = y[c] - mu; var += t * t; }
    var *= (1.0f / 16.0f);
    float inv = 1.0f / sqrtf(var + 1e-5f);

    float o[16];
#pragma unroll
    for (int c = 0; c < 16; ++c)
      o[c] = (y[c] - mu) * inv * lds_par[16 + c] + lds_par[32 + c];

    const long long ob = tok * 16;
#pragma unroll
    for (int q = 0; q < 4; ++q) {
      float4 v; v.x = o[q*4+0]; v.y = o[q*4+1]; v.z = o[q*4+2]; v.w = o[q*4+3];
      *(float4*)&out_x[ob + q * 4] = v;    // 64B/lane, fully coalesced
    }
    float2 av; av.x = a0; av.y = a1;
    *(float2*)&out_a[tok * 2] = av;
  }
}

extern "C" void kernel_launch(void* const* d_in, const int* in_sizes, int n_in,
                              void* d_out, int out_size, void* d_ws, size_t ws_size,
                              hipStream_t stream) {
  (void)n_in; (void)out_size; (void)d_ws; (void)ws_size;
  const float* x      = (const float*)d_in[0];
  const float* conv_w = (const float*)d_in[1];
  const float* conv_b = (const float*)d_in[2];
  const float* w1     = (const float*)d_in[3];
  const float* b1     = (const float*)d_in[4];
  const float* w2     = (const float*)d_in[5];
  const float* b2     = (const float*)d_in[6];
  const float* ln_g   = (const float*)d_in[7];
  const float* ln_b   = (const float*)d_in[8];

  const long long n_tokens = (long long)in_sizes[0] / 16;  // B*T
  float* out_x = (float*)d_out;
  float* out_a = out_x + (size_t)in_sizes[0];              // alphas follow out_X

  dim3 grid((unsigned)(n_tokens / CHUNK));                 // 8192 blocks
  dim3 block(256);
  bimanual_fused<<<grid, block, 0, stream>>>(x, conv_w, conv_b, w1, b1, w2, b2,
                                             ln_g, ln_b, out_x, out_a);
}